// WalkerIntegrator_82214263980492
// MI455X (gfx1250) — compile-verified
//
#include <hip/hip_runtime.h>
#include <math.h>

// ---------------------------------------------------------------------------
// WalkerIntegrator: out[0] = x0; out[1+t] = (x0.xy + cumsum(tanh(U)*vmax*dt),
//                                            atan2(vy, vx+1e-9), clip(|v|,0,vmax))
// 3-pass deterministic scan. H = 2^24: U (134MB) fits MI455X's 192MB L2, so the
// pass-C re-read of U is an L2 hit; HBM traffic ~= 402MB floor (~17us @23.3TB/s).
// Intra-wave 256-element scan runs on the matrix pipe via V_WMMA_F32_16X16X4_F32:
//   Y = L*X (4 chained K-chunks), D = T*SU + Y  (T = column totals, SU strict-upper).
// ---------------------------------------------------------------------------

typedef __attribute__((ext_vector_type(2))) float v2f;
typedef __attribute__((ext_vector_type(8))) float v8f;

#define VMAX 2.2f
#define TILE 8192              // timesteps per block (kernels A and C)
#define BLOCK_THREADS 256
#define WAVES 8
#define WAVE_TILE 256          // timesteps scanned per wave per iteration
#define ITERS (TILE / (WAVES * WAVE_TILE))   // = 4

__device__ __forceinline__ v8f wmma_f32_16x16x4(v2f a, v2f b, v8f c) {
  // 8 args: (neg_a, A, neg_b, B, c_mod, C, reuse_a, reuse_b)
  return __builtin_amdgcn_wmma_f32_16x16x4_f32(false, a, false, b, (short)0, c,
                                               false, false);
}

// --------------------------- Pass A: block sums -----------------------------
__global__ void __launch_bounds__(BLOCK_THREADS)
walker_partial_sums(const float* __restrict__ U, const float* __restrict__ dtp,
                    float2* __restrict__ blockSums, int H) {
  const int tid = threadIdx.x;
  const int bb  = blockIdx.x;
  const float dt = *dtp;
  const int perThread = TILE / BLOCK_THREADS;   // 32 timesteps
  int base = bb * TILE + tid * perThread;
  float sx = 0.f, sy = 0.f;
  #pragma unroll 4
  for (int i = 0; i < perThread; i += 2) {
    int t = base + i;
    if (t + 1 < H) {
      float4 q = *(const float4*)(U + 2 * t);   // two timesteps, both channels
      sx += tanhf(q.x); sy += tanhf(q.y);
      sx += tanhf(q.z); sy += tanhf(q.w);
    }
  }
  // wave32 reduction, fixed order -> deterministic
  #pragma unroll
  for (int off = 16; off > 0; off >>= 1) {
    sx += __shfl_xor(sx, off, 32);
    sy += __shfl_xor(sy, off, 32);
  }
  __shared__ float wx[WAVES], wy[WAVES];
  if ((tid & 31) == 0) { wx[tid >> 5] = sx; wy[tid >> 5] = sy; }
  __syncthreads();
  if (tid == 0) {
    float tx = 0.f, ty = 0.f;
    for (int w = 0; w < WAVES; ++w) { tx += wx[w]; ty += wy[w]; }
    blockSums[bb] = make_float2(tx * (VMAX * dt), ty * (VMAX * dt));
  }
}

// ------------------- Pass B: exclusive scan of block sums -------------------
__global__ void __launch_bounds__(1024)
walker_scan_blocks(const float2* __restrict__ sums, float2* __restrict__ excl,
                   const float* __restrict__ x0, float* __restrict__ out, int nb) {
  const int tid = threadIdx.x;                  // 1024 threads
  const int npt = (nb + 1023) >> 10;            // entries per thread (contiguous)
  float tx = 0.f, ty = 0.f;
  for (int i = 0; i < npt; ++i) {
    int idx = tid * npt + i;
    if (idx < nb) { float2 s = sums[idx]; tx += s.x; ty += s.y; }
  }
  __shared__ float bx[1024], by[1024];
  bx[tid] = tx; by[tid] = ty;
  for (int off = 1; off < 1024; off <<= 1) {    // Hillis-Steele inclusive scan
    __syncthreads();
    float ax = (tid >= off) ? bx[tid - off] : 0.f;
    float ay = (tid >= off) ? by[tid - off] : 0.f;
    __syncthreads();
    bx[tid] += ax; by[tid] += ay;
  }
  __syncthreads();
  float ox = (tid > 0) ? bx[tid - 1] : 0.f;     // exclusive thread offset
  float oy = (tid > 0) ? by[tid - 1] : 0.f;
  for (int i = 0; i < npt; ++i) {
    int idx = tid * npt + i;
    if (idx < nb) {
      excl[idx] = make_float2(ox, oy);
      float2 s = sums[idx];
      ox += s.x; oy += s.y;
    }
  }
  if (tid == 0) {                               // output row 0 = x0
    out[0] = x0[0]; out[1] = x0[1]; out[2] = x0[2]; out[3] = x0[3];
  }
}

// --------------- Pass C: WMMA intra-wave scan + emit trajectory -------------
__global__ void __launch_bounds__(BLOCK_THREADS)
walker_scan_emit(const float* __restrict__ U, const float* __restrict__ x0p,
                 const float* __restrict__ dtp, const float2* __restrict__ excl,
                 float* __restrict__ out, int H) {
  const int tid  = threadIdx.x, bb = blockIdx.x;
  const int wave = tid >> 5, ln = tid & 31;
  const int n    = ln & 15;          // column / A-row within 16x16 tile
  const int hi   = ln >> 4;          // half-wave select (K offset 2*hi)
  const float dt = *dtp;
  const float vmax = VMAX;
  float2 be = excl[bb];
  float carryx = x0p[0] + be.x;
  float carryy = x0p[1] + be.y;
  __shared__ float wsx[WAVES], wsy[WAVES];

  for (int it = 0; it < ITERS; ++it) {
    int waveBase = bb * TILE + it * (WAVES * WAVE_TILE) + wave * WAVE_TILE;

    // Load: lane (n,hi) supplies B-chunk values X[k,n] = s[16n+k], k=4c+2hi+{0,1}.
    // Pairs are consecutive timesteps -> one b128 load covers both channels.
    v2f uvx[4], uvy[4];
    #pragma unroll
    for (int c = 0; c < 4; ++c) {
      int t = waveBase + 16 * n + 4 * c + 2 * hi;
      float4 q = *(const float4*)(U + 2 * t);   // (Ux[t],Uy[t],Ux[t+1],Uy[t+1])
      uvx[c].x = vmax * tanhf(q.x); uvy[c].x = vmax * tanhf(q.y);
      uvx[c].y = vmax * tanhf(q.z); uvy[c].y = vmax * tanhf(q.w);
    }

    // Chain 1: Y = L * X  (L[m,k] = k<=m), per channel, 4 K-chunks.
    v8f accx = {0.f,0.f,0.f,0.f,0.f,0.f,0.f,0.f};
    v8f accy = {0.f,0.f,0.f,0.f,0.f,0.f,0.f,0.f};
    #pragma unroll
    for (int c = 0; c < 4; ++c) {
      int k0 = 4 * c + 2 * hi;
      v2f a; a.x = (k0 <= n) ? 1.f : 0.f; a.y = (k0 + 1 <= n) ? 1.f : 0.f;
      v2f bxv; bxv.x = uvx[c].x * dt; bxv.y = uvx[c].y * dt;
      v2f byv; byv.x = uvy[c].x * dt; byv.y = uvy[c].y * dt;
      accx = wmma_f32_16x16x4(a, bxv, accx);
      accy = wmma_f32_16x16x4(a, byv, accy);
    }

    // Column totals t[j] = Y[15,j] live in lane 16+j of acc*[7].
    // Chain 2: D = T*SU + Y, T[m,k]=t[k], SU[k,n] = k<n  -> full 256-elem scan.
    v8f dx = accx, dy = accy;
    #pragma unroll
    for (int c = 0; c < 4; ++c) {
      int k0 = 4 * c + 2 * hi;
      v2f ax, ay, b;
      ax.x = __shfl(accx[7], 16 + k0,     32);
      ax.y = __shfl(accx[7], 16 + k0 + 1, 32);
      ay.x = __shfl(accy[7], 16 + k0,     32);
      ay.y = __shfl(accy[7], 16 + k0 + 1, 32);
      b.x = (k0     < n) ? 1.f : 0.f;
      b.y = (k0 + 1 < n) ? 1.f : 0.f;
      dx = wmma_f32_16x16x4(ax, b, dx);
      dy = wmma_f32_16x16x4(ay, b, dy);
    }

    // Wave total = scan value of element 255 = lane 31, reg 7.
    float wtx = __shfl(dx[7], 31, 32);
    float wty = __shfl(dy[7], 31, 32);
    if (ln == 0) { wsx[wave] = wtx; wsy[wave] = wty; }
    __syncthreads();
    float offx = carryx, offy = carryy, itx = 0.f, ity = 0.f;
    for (int w = 0; w < WAVES; ++w) {           // fixed order, all lanes agree
      if (w < wave) { offx += wsx[w]; offy += wsy[w]; }
      itx += wsx[w]; ity += wsy[w];
    }
    __syncthreads();
    carryx += itx; carryy += ity;

    // Emit: reg r <-> element k = 8*hi + r of column n. uv for element k lives
    // in lane with hi' = (k>>1)&1 = (r>>1)&1; exchange across halves via
    // shfl_xor(16). All register indices compile-time (no scratch spills).
    #pragma unroll
    for (int r = 0; r < 8; ++r) {
      const int rA  = r >> 2;        // low-half chunk index
      const int j   = r & 1;
      const int bit = (r >> 1) & 1;  // owning half-wave of element k
      float Lx = j ? uvx[rA].y     : uvx[rA].x;
      float Hx = j ? uvx[2 + rA].y : uvx[2 + rA].x;
      float Ly = j ? uvy[rA].y     : uvy[rA].x;
      float Hy = j ? uvy[2 + rA].y : uvy[2 + rA].x;
      float localx = hi ? Hx : Lx, sendx = hi ? Lx : Hx;
      float localy = hi ? Hy : Ly, sendy = hi ? Ly : Hy;
      float px_ = __shfl_xor(sendx, 16, 32);
      float py_ = __shfl_xor(sendy, 16, 32);
      float vx = (hi == bit) ? localx : px_;
      float vy = (hi == bit) ? localy : py_;

      float posx = dx[r] + offx;
      float posy = dy[r] + offy;
      float th   = atan2f(vy, vx + 1e-9f);
      float sp   = fminf(sqrtf(vx * vx + vy * vy + 1e-12f), vmax);

      int t = waveBase + 16 * n + 8 * hi + r;
      *(float4*)(out + 4 * (t + 1)) = make_float4(posx, posy, th, sp);
    }
  }
}

// ------------------------------- launcher -----------------------------------
extern "C" void kernel_launch(void* const* d_in, const int* in_sizes, int n_in,
                              void* d_out, int out_size, void* d_ws, size_t ws_size,
                              hipStream_t stream) {
  const float* x0 = (const float*)d_in[0];
  const float* U  = (const float*)d_in[1];
  const float* dt = (const float*)d_in[2];
  float* out = (float*)d_out;
  int H  = in_sizes[1] / 2;          // 16777216
  int nb = H / TILE;                 // 2048 blocks (H multiple of 8192)

  float2* sums = (float2*)d_ws;      // nb float2
  float2* excl = sums + nb;          // nb float2   (32 KB total)

  walker_partial_sums<<<nb, BLOCK_THREADS, 0, stream>>>(U, dt, sums, H);
  walker_scan_blocks<<<1, 1024, 0, stream>>>(sums, excl, x0, out, nb);
  walker_scan_emit<<<nb, BLOCK_THREADS, 0, stream>>>(U, x0, dt, excl, out, H);
}